// SimpleMatcher_19868518711419
// MI455X (gfx1250) — compile-verified
//
#include <hip/hip_runtime.h>
#include <hip/hip_bf16.h>

#define BATCH 64
#define NGT   64
#define NANCH 8432   // 5476 + 2166 + 600 + 150 + 36 + 4

// ---------------------------------------------------------------------------
// Kernel 1: decode one SSD level: (cx,cy,h,w) -> xyxy into pred[B][N][4]
// ---------------------------------------------------------------------------
__global__ void decode_level_kernel(const float* __restrict__ t,   // (B, count, 4)
                                    const float* __restrict__ d,   // (count, 4)
                                    float* __restrict__ pred,      // (B, NANCH, 4)
                                    int count, int level_off) {
    int nl = blockIdx.x * blockDim.x + threadIdx.x;
    int b  = blockIdx.y;
    if (nl >= count) return;
    float4 tv = reinterpret_cast<const float4*>(t)[(size_t)b * count + nl];
    float4 dv = reinterpret_cast<const float4*>(d)[nl];
    float cx = dv.x + dv.w * tv.x;          // d_cx + d_w * t0
    float cy = dv.y + dv.z * tv.y;          // d_cy + d_h * t1
    float h  = dv.z * __expf(tv.z);
    float w  = dv.w * __expf(tv.w);
    float4 o;
    o.x = cx - 0.5f * w;
    o.y = cy - 0.5f * h;
    o.z = cx + 0.5f * w;
    o.w = cy + 0.5f * h;
    reinterpret_cast<float4*>(pred)[(size_t)b * NANCH + level_off + nl] = o;
}

// ---------------------------------------------------------------------------
// Kernel 2: GT boxes cxcyhw -> xyxy + areas
// ---------------------------------------------------------------------------
__global__ void gt_convert_kernel(const float* __restrict__ gt,    // (B, NGT, 4)
                                  float4* __restrict__ gtXY,       // (B*NGT)
                                  float* __restrict__ areaG) {     // (B*NGT)
    int i = blockIdx.x * blockDim.x + threadIdx.x;
    if (i >= BATCH * NGT) return;
    float4 v = reinterpret_cast<const float4*>(gt)[i];
    float4 o;
    o.x = v.x - 0.5f * v.w;
    o.y = v.y - 0.5f * v.z;
    o.z = v.x + 0.5f * v.w;
    o.w = v.y + 0.5f * v.z;
    gtXY[i]  = o;
    areaG[i] = (o.z - o.x) * (o.w - o.y);
}

__device__ __forceinline__ float iou_xyxy(float4 p, float ap, float4 g, float ag) {
    float ltx = fmaxf(p.x, g.x), lty = fmaxf(p.y, g.y);
    float rbx = fminf(p.z, g.z), rby = fminf(p.w, g.w);
    float iw = fmaxf(rbx - ltx, 0.0f);
    float ih = fmaxf(rby - lty, 0.0f);
    float inter = iw * ih;
    float uni = fmaxf(ap + ag - inter, 1e-9f);
    return inter / uni;
}

// ---------------------------------------------------------------------------
// Kernel 3: per-(b,g) argmax of IoU over all NANCH anchors.
// One 256-thread block per (g, b); first-index tie-break like jnp.argmax.
// ---------------------------------------------------------------------------
__global__ void argmax_kernel(const float4* __restrict__ predXY,
                              const float4* __restrict__ gtXY,
                              const float*  __restrict__ areaG,
                              int* __restrict__ maxInd) {
    int g = blockIdx.x;
    int b = blockIdx.y;
    float4 gv = gtXY[b * NGT + g];
    float  ag = areaG[b * NGT + g];
    const float4* pp = predXY + (size_t)b * NANCH;

    float bestV = -1.0f;
    int   bestN = 0;
    for (int n = threadIdx.x; n < NANCH; n += 256) {
        float4 p = pp[n];
        float ap = (p.z - p.x) * (p.w - p.y);
        float iou = iou_xyxy(p, ap, gv, ag);
        if (iou > bestV) { bestV = iou; bestN = n; }   // ascending n => first max
    }

    __shared__ float sv[256];
    __shared__ int   si[256];
    sv[threadIdx.x] = bestV;
    si[threadIdx.x] = bestN;
    __syncthreads();
    for (int s = 128; s > 0; s >>= 1) {
        if (threadIdx.x < s) {
            float v2 = sv[threadIdx.x + s];
            int   i2 = si[threadIdx.x + s];
            if (v2 > sv[threadIdx.x] ||
                (v2 == sv[threadIdx.x] && i2 < si[threadIdx.x])) {
                sv[threadIdx.x] = v2;
                si[threadIdx.x] = i2;
            }
        }
        __syncthreads();
    }
    if (threadIdx.x == 0) maxInd[b * NGT + g] = si[0];
}

// ---------------------------------------------------------------------------
// Kernel 4: fused match / pos / neg writer.
// Per-batch GT state (1 KB boxes + 256 B areas + 256 B argmax idx) is staged
// into LDS with gfx1250 async global->LDS b128 loads (ASYNCcnt path).
// ---------------------------------------------------------------------------
__device__ __forceinline__ void async_to_lds_b128(void* lds_ptr, const void* gptr) {
    // Low 32 bits of a generic pointer to LDS are the LDS byte address
    // (aperture check discards upper bits).
    unsigned int lds_off = (unsigned int)(uintptr_t)lds_ptr;
    asm volatile("global_load_async_to_lds_b128 %0, %1, off"
                 :: "v"(lds_off), "v"(gptr)
                 : "memory");
}

__global__ void match_kernel(const float4* __restrict__ predXY,
                             const float4* __restrict__ gtXY,
                             const float*  __restrict__ areaG,
                             const int*    __restrict__ maxInd,
                             float* __restrict__ out) {
    __shared__ float4 sGT[NGT];    // 1024 B
    __shared__ float  sAG[NGT];    //  256 B
    __shared__ int    sIdx[NGT];   //  256 B

    const int b   = blockIdx.y;
    const int tid = threadIdx.x;

    // 96 lanes each issue one async 16B global->LDS transfer.
    if (tid < 64) {
        async_to_lds_b128(&sGT[tid], gtXY + b * NGT + tid);
    } else if (tid < 80) {
        int c = tid - 64;
        async_to_lds_b128(&sAG[c * 4], areaG + b * NGT + c * 4);
    } else if (tid < 96) {
        int c = tid - 80;
        async_to_lds_b128(&sIdx[c * 4], maxInd + b * NGT + c * 4);
    }
    asm volatile("s_wait_asynccnt 0" ::: "memory");
    __syncthreads();

    const int n = blockIdx.x * blockDim.x + tid;
    if (n >= NANCH) return;

    float4 p = predXY[(size_t)b * NANCH + n];
    float ap = (p.z - p.x) * (p.w - p.y);

    float* mrow = out + ((size_t)b * NANCH + n) * NGT;
    bool any = false;
#pragma unroll 4
    for (int g0 = 0; g0 < NGT; g0 += 4) {
        float4 m;
        float mv[4];
#pragma unroll
        for (int j = 0; j < 4; ++j) {
            int g = g0 + j;
            float iou = iou_xyxy(p, ap, sGT[g], sAG[g]);
            bool forced = (sIdx[g] == n);
            bool hit = forced || (iou >= 0.5f);
            any |= hit;
            mv[j] = hit ? 1.0f : 0.0f;
        }
        m.x = mv[0]; m.y = mv[1]; m.z = mv[2]; m.w = mv[3];
        reinterpret_cast<float4*>(mrow)[g0 >> 2] = m;   // 256B/thread, coalesced
    }

    const size_t MATCH = (size_t)BATCH * NANCH * NGT;
    const size_t PN    = (size_t)BATCH * NANCH;
    out[MATCH + (size_t)b * NANCH + n]      = any ? 1.0f : 0.0f;
    out[MATCH + PN + (size_t)b * NANCH + n] = any ? 0.0f : 1.0f;
}

// ---------------------------------------------------------------------------
extern "C" void kernel_launch(void* const* d_in, const int* in_sizes, int n_in,
                              void* d_out, int out_size, void* d_ws, size_t ws_size,
                              hipStream_t stream) {
    (void)in_sizes; (void)n_in; (void)out_size; (void)ws_size;

    static const int counts[6] = {5476, 2166, 600, 150, 36, 4};
    static const int offs[6]   = {0, 5476, 7642, 8242, 8392, 8428};

    // Workspace layout (all 16B-aligned):
    float* predXY = (float*)d_ws;                                  // B*N*4 floats
    float* gtXYf  = predXY + (size_t)BATCH * NANCH * 4;            // B*64*4
    float* areaG  = gtXYf + BATCH * NGT * 4;                       // B*64
    int*   maxInd = (int*)(areaG + BATCH * NGT);                   // B*64

    // 1) decode each level
    for (int l = 0; l < 6; ++l) {
        dim3 grid((counts[l] + 255) / 256, BATCH);
        decode_level_kernel<<<grid, 256, 0, stream>>>(
            (const float*)d_in[l], (const float*)d_in[7 + l],
            predXY, counts[l], offs[l]);
    }

    // 2) gt convert
    gt_convert_kernel<<<(BATCH * NGT + 255) / 256, 256, 0, stream>>>(
        (const float*)d_in[6], (float4*)gtXYf, areaG);

    // 3) argmax over anchors per (b,g)
    argmax_kernel<<<dim3(NGT, BATCH), 256, 0, stream>>>(
        (const float4*)predXY, (const float4*)gtXYf, areaG, maxInd);

    // 4) fused match/pos/neg
    match_kernel<<<dim3((NANCH + 255) / 256, BATCH), 256, 0, stream>>>(
        (const float4*)predXY, (const float4*)gtXYf, areaG, maxInd,
        (float*)d_out);
}